// CVXLoss_41841571397808
// MI455X (gfx1250) — compile-verified
//
#include <hip/hip_runtime.h>

#define BATCH   8192
#define NCLS    8192
#define KVAL    5.0f
#define NBIS    60
#define BLOCK   512
#define PER_TH  16          // NCLS / BLOCK
#define NWAVES  16          // BLOCK / 32
#define LOG_N   9.010913347279288f   // log(8192)

typedef __attribute__((ext_vector_type(2))) float v2f;
typedef __attribute__((ext_vector_type(8))) float v8f;

__device__ __forceinline__ float wave_sum(float v) {
#pragma unroll
  for (int m = 16; m >= 1; m >>= 1) v += __shfl_xor(v, m, 32);
  return v;
}
__device__ __forceinline__ float wave_min(float v) {
#pragma unroll
  for (int m = 16; m >= 1; m >>= 1) v = fminf(v, __shfl_xor(v, m, 32));
  return v;
}
__device__ __forceinline__ float wave_max(float v) {
#pragma unroll
  for (int m = 16; m >= 1; m >>= 1) v = fmaxf(v, __shfl_xor(v, m, 32));
  return v;
}

__device__ __forceinline__ float block_sum(float v, float* sred, int wid, int lane) {
  v = wave_sum(v);
  if (lane == 0) sred[wid] = v;
  __syncthreads();
  float t = 0.f;
#pragma unroll
  for (int k = 0; k < NWAVES; ++k) t += sred[k];
  __syncthreads();
  return t;
}
__device__ __forceinline__ float block_min(float v, float* sred, int wid, int lane) {
  v = wave_min(v);
  if (lane == 0) sred[wid] = v;
  __syncthreads();
  float t = sred[0];
#pragma unroll
  for (int k = 1; k < NWAVES; ++k) t = fminf(t, sred[k]);
  __syncthreads();
  return t;
}
__device__ __forceinline__ float block_max(float v, float* sred, int wid, int lane) {
  v = wave_max(v);
  if (lane == 0) sred[wid] = v;
  __syncthreads();
  float t = sred[0];
#pragma unroll
  for (int k = 1; k < NWAVES; ++k) t = fmaxf(t, sred[k]);
  __syncthreads();
  return t;
}

__global__ __launch_bounds__(BLOCK)
void cvx_row_kernel(const float* __restrict__ x, const int* __restrict__ y,
                    float* __restrict__ rowloss) {
  __shared__ __align__(16) float srow[NCLS];     // 32 KB row stage (of 320 KB/WGP)
  __shared__ __align__(16) float sred[NWAVES];   // wave partials, read back as float4

  const int t    = threadIdx.x;
  const int lane = t & 31;
  const int wid  = t >> 5;
  const int row  = blockIdx.x;
  const float* __restrict__ xr = x + (size_t)row * NCLS;

  // ---- CDNA5 async copy: row -> LDS via GLOBAL_LOAD_ASYNC_TO_LDS_B128 (ASYNCcnt) ----
#pragma unroll
  for (int jj = 0; jj < 4; ++jj) {
    const int i4 = t + BLOCK * jj;   // float4 index, fully coalesced
    unsigned lofs = (unsigned)(unsigned long long)(srow + i4 * 4);
    unsigned long long ga = (unsigned long long)(xr + i4 * 4);
    asm volatile("global_load_async_to_lds_b128 %0, %1, off"
                 :: "v"(lofs), "v"(ga) : "memory");
  }
  asm volatile("s_wait_asynccnt 0" ::: "memory");

  // ---- pull this thread's 16 elements from LDS; sum of squares ----
  float w[PER_TH];
  float ss = 0.f;
#pragma unroll
  for (int jj = 0; jj < 4; ++jj) {
    const int i4 = t + BLOCK * jj;
    float4 q = ((const float4*)srow)[i4];
    w[4*jj+0] = q.x; w[4*jj+1] = q.y; w[4*jj+2] = q.z; w[4*jj+3] = q.w;
    ss += q.x*q.x + q.y*q.y + q.z*q.z + q.w*q.w;
  }
  const float sumsq = block_sum(ss, sred, wid, lane);
  const float inv   = 1.0f / fmaxf(sqrtf(sumsq), 1e-12f);   // TAU = 1

  // ---- e = xn / TAU ; precompute w = exp(e) once; track min/max of e ----
  float mn =  1e30f, mx = -1e30f;
#pragma unroll
  for (int j = 0; j < PER_TH; ++j) {
    const float ej = w[j] * inv;
    mn = fminf(mn, ej);
    mx = fmaxf(mx, ej);
    w[j] = expf(ej);                  // exp factored out of the bisection loop
  }
  const float emin = block_min(mn, sred, wid, lane);
  const float emax = block_max(mx, sred, wid, lane);

  float lo = emin - 1.0f;             // EPS = 1
  float hi = emax + LOG_N;            // EPS * log(n)

  v2f bones; bones[0] = 1.0f; bones[1] = 1.0f;   // B (4x16) = all ones

  // ---- 60 bisection iterations: f(mid) = sum min(1, w_i * exp(-mid-1)) ----
  for (int it = 0; it < NBIS; ++it) {
    const float mid = 0.5f * (lo + hi);
    const float s   = expf(-mid - 1.0f);         // uniform -> SALU v_s_exp_f32

    // two independent accumulating WMMA chains (break the D->C RAW chain)
    v8f acc0 = {0.f, 0.f, 0.f, 0.f, 0.f, 0.f, 0.f, 0.f};
    v8f acc1 = {0.f, 0.f, 0.f, 0.f, 0.f, 0.f, 0.f, 0.f};
#pragma unroll
    for (int jj = 0; jj < 4; ++jj) {
      v2f a0, a1;
      a0[0] = fminf(w[4*jj + 0] * s, 1.0f);
      a0[1] = fminf(w[4*jj + 1] * s, 1.0f);
      a1[0] = fminf(w[4*jj + 2] * s, 1.0f);
      a1[1] = fminf(w[4*jj + 3] * s, 1.0f);
      acc0 = __builtin_amdgcn_wmma_f32_16x16x4_f32(false, a0, false, bones,
                                                   (short)0, acc0, false, false);
      acc1 = __builtin_amdgcn_wmma_f32_16x16x4_f32(false, a1, false, bones,
                                                   (short)0, acc1, false, false);
    }
    // D[m][n] = rowsum[m]; fold 8 VGPRs per lane, then one xor-16 shuffle
    // yields the 512-element wave total in every lane.
    float part = (acc0[0] + acc1[0]) + (acc0[1] + acc1[1]) +
                 (acc0[2] + acc1[2]) + (acc0[3] + acc1[3]) +
                 (acc0[4] + acc1[4]) + (acc0[5] + acc1[5]) +
                 (acc0[6] + acc1[6]) + (acc0[7] + acc1[7]);
    part += __shfl_xor(part, 16, 32);

    sred[wid] = part;                 // branchless: all 32 lanes write same value
    __syncthreads();
    float4 r0 = ((const float4*)sred)[0];
    float4 r1 = ((const float4*)sred)[1];
    float4 r2 = ((const float4*)sred)[2];
    float4 r3 = ((const float4*)sred)[3];
    const float total = ((r0.x + r0.y) + (r0.z + r0.w)) +
                        ((r1.x + r1.y) + (r1.z + r1.w)) +
                        ((r2.x + r2.y) + (r2.z + r2.w)) +
                        ((r3.x + r3.y) + (r3.z + r3.w));
    __syncthreads();

    if (total > KVAL) lo = mid; else hi = mid;
  }

  // ---- differentiable-Newton step on the dual residual (forward value only) ----
  const float nu0 = 0.5f * (lo + hi);
  const float s0  = expf(-nu0 - 1.0f);
  float gp = 0.f, gq = 0.f;
#pragma unroll
  for (int j = 0; j < PER_TH; ++j) {
    const float ws = w[j] * s0;
    gp += fminf(ws, 1.0f);
    gq += (ws < 1.0f) ? ws : 0.f;
  }
  const float gsum   = block_sum(gp, sred, wid, lane);
  const float gnusum = block_sum(gq, sred, wid, lane);
  const float g      = gsum - KVAL;
  const float g_nu   = -gnusum;            // / EPS
  const float nu     = nu0 - g / g_nu;

  if (t == 0) {
    const int yc   = y[row];
    const float ey = srow[yc] * inv;       // raw x still staged in LDS
    const float p  = fminf(1.0f, expf(ey - nu - 1.0f));
    rowloss[row]   = -logf(p + 1e-8f);
  }
}

__global__ __launch_bounds__(256)
void cvx_mean_kernel(const float* __restrict__ rl, float* __restrict__ out) {
  __shared__ float sred[8];
  const int t = threadIdx.x;
  float v = 0.f;
  for (int i = t; i < BATCH; i += 256) v += rl[i];
  v = wave_sum(v);
  if ((t & 31) == 0) sred[t >> 5] = v;
  __syncthreads();
  if (t == 0) {
    float s = 0.f;
#pragma unroll
    for (int k = 0; k < 8; ++k) s += sred[k];
    out[0] = s * (1.0f / (float)BATCH);
  }
}

extern "C" void kernel_launch(void* const* d_in, const int* in_sizes, int n_in,
                              void* d_out, int out_size, void* d_ws, size_t ws_size,
                              hipStream_t stream) {
  const float* x = (const float*)d_in[0];
  const int*   y = (const int*)d_in[1];
  float* rowloss = (float*)d_ws;           // BATCH floats of scratch

  cvx_row_kernel<<<BATCH, BLOCK, 0, stream>>>(x, y, rowloss);
  cvx_mean_kernel<<<1, 256, 0, stream>>>(rowloss, (float*)d_out);
}